// Critic_13469017441090
// MI455X (gfx1250) — compile-verified
//
#include <hip/hip_runtime.h>

// GAE reverse scan: returns[t] = (rewards[t] + DV*values[t+1]) + COEF*returns[t+1]
#define COEF (0.99f * 0.95f)
#define DVAL (0.99f * (1.0f - 0.95f))

constexpr int CHUNK  = 32;   // time steps per wave-chunk (register-resident results)
constexpr int NCHUNK = 32;   // chunks per block == waves per block (threadIdx.y)
constexpr int COLS   = 32;   // columns per block == wave32 lanes (threadIdx.x)
constexpr int GROUP  = 8;    // timesteps per async-staged subtile (double-buffered)
constexpr int NGROUP = CHUNK / GROUP;  // 4 groups

typedef int v4i __attribute__((vector_size(16)));        // int4 for b128 builtin
typedef __attribute__((address_space(1))) int gint_as;   // global ("__device__") int
typedef __attribute__((address_space(3))) int lint_as;   // LDS ("__shared__") int
typedef __attribute__((address_space(1))) v4i gv4i_as;   // global int4
typedef __attribute__((address_space(3))) v4i lv4i_as;   // LDS int4
typedef const __attribute__((address_space(1))) float* gfp_t;

#if __has_builtin(__builtin_amdgcn_global_load_async_to_lds_b128)
#define HAVE_ASYNC_B128 1
#else
#define HAVE_ASYNC_B128 0
#endif

// ops per group (both streams): b128 moves 4 rows x 32 cols (512B) per op.
constexpr int OPS_PER_GROUP = HAVE_ASYNC_B128 ? 2 * (GROUP / 4) : 2 * GROUP;

// --- CDNA5 async global->LDS copies (ASYNCcnt-tracked) ---
__device__ __forceinline__ void async_copy_f32(gfp_t src, float* dst_generic) {
#if __has_builtin(__builtin_amdgcn_global_load_async_to_lds_b32)
    __builtin_amdgcn_global_load_async_to_lds_b32(
        (gint_as*)src, (lint_as*)dst_generic, 0, 0);
#else
    unsigned laddr = (unsigned)(unsigned long long)(lint_as*)dst_generic;
    asm volatile("global_load_async_to_lds_b32 %0, %1, off"
                 :: "v"(laddr), "v"(src) : "memory");
#endif
}

#if HAVE_ASYNC_B128
__device__ __forceinline__ void async_copy_b128(gfp_t src, float* dst_generic) {
    __builtin_amdgcn_global_load_async_to_lds_b128(
        (gv4i_as*)src, (lv4i_as*)dst_generic, 0, 0);
}
#endif

template <int N>
__device__ __forceinline__ void wait_asynccnt() {
#if __has_builtin(__builtin_amdgcn_s_wait_asynccnt)
    __builtin_amdgcn_s_wait_asynccnt(N);
#else
    asm volatile("s_wait_asynccnt %0" :: "n"(N) : "memory");
#endif
}

__device__ __forceinline__ void wait_dscnt0() {
#if __has_builtin(__builtin_amdgcn_s_wait_dscnt)
    __builtin_amdgcn_s_wait_dscnt(0);
#else
    asm volatile("s_wait_dscnt 0x0" ::: "memory");
#endif
}

// Fast path: T == 1024, B compile-time constant (immediate-offset addressing).
template <int B>
__global__ __launch_bounds__(NCHUNK * COLS) void gae_fast(
    const float* __restrict__ rewards,
    const float* __restrict__ values,
    float* __restrict__ out)
{
    // [wave][buf][stream][k][lane] : 32*2*2*8*32*4 = 128 KB LDS (<= 320 KB/WGP).
    __shared__ float stage[NCHUNK][2][2][GROUP][COLS];
    // Carry slots alias each wave's own stage region (written only after that
    // region is fully consumed; readers separated by __syncthreads()).
#define CARRY(c, l) stage[c][0][0][0][l]

    const int lane  = threadIdx.x;
    const int chunk = threadIdx.y;
    const int col0  = blockIdx.x * COLS;       // block's first column (128B aligned)
    const int col   = col0 + lane;
    const int t0    = chunk * CHUNK;

    gfp_t rp = (gfp_t)(rewards + (size_t)t0 * B + col);        // rewards[t0 + i]
    gfp_t vp = (gfp_t)(values  + (size_t)(t0 + 1) * B + col);  // values[t0 + 1 + i]

#if HAVE_ASYNC_B128
    // b128 lane mapping: lane -> (subrow = lane>>3, 16B column slice = (lane&7)*4)
    const int subrow = lane >> 3;
    const int subcol = (lane & 7) * 4;
    gfp_t rp128 = (gfp_t)(rewards + ((size_t)t0 + subrow) * B + col0 + subcol);
    gfp_t vp128 = (gfp_t)(values  + ((size_t)t0 + 1 + subrow) * B + col0 + subcol);
#endif

    auto issue = [&](int g, int buf) {
#if HAVE_ASYNC_B128
#pragma unroll
        for (int m = 0; m < GROUP / 4; ++m) {
            const size_t roff = (size_t)(g * GROUP + m * 4) * B;
            async_copy_b128(rp128 + roff, &stage[chunk][buf][0][m * 4 + subrow][subcol]);
            async_copy_b128(vp128 + roff, &stage[chunk][buf][1][m * 4 + subrow][subcol]);
        }
#else
#pragma unroll
        for (int k = 0; k < GROUP; ++k) {
            const int i = g * GROUP + k;
            async_copy_f32(rp + (size_t)i * B, &stage[chunk][buf][0][k][lane]);
            async_copy_f32(vp + (size_t)i * B, &stage[chunk][buf][1][k][lane]);
        }
#endif
    };

    // Prime the double-buffered async pipeline (2 groups = 4 KB/wave in flight).
    issue(NGROUP - 1, (NGROUP - 1) & 1);
    issue(NGROUP - 2, (NGROUP - 2) & 1);

    // Phase A: chunk-local reverse scan; results stay in VGPRs.
    float local[CHUNK];
    float r = 0.0f;
#pragma unroll
    for (int g = NGROUP - 1; g >= 0; --g) {
        if (g > 0) wait_asynccnt<OPS_PER_GROUP>();  // oldest group retired (in-order)
        else       wait_asynccnt<0>();
        const int buf = g & 1;
#pragma unroll
        for (int k = GROUP - 1; k >= 0; --k) {
            const int i = g * GROUP + k;
            float rw = stage[chunk][buf][0][k][lane];
            float vv = stage[chunk][buf][1][k][lane];
            r = (rw + DVAL * vv) + COEF * r;
            local[i] = r;
        }
        if (g >= 2) {
            wait_dscnt0();          // async LDS writes are NOT ordered vs ds_loads
            issue(g - 2, g & 1);    // refill the buffer we just consumed
        }
    }

    // Publish chunk carry (local value at chunk start) and sync the workgroup.
    CARRY(chunk, lane) = r;
    __syncthreads();

    // coef^CHUNK, constant-folded.
    float coefL = 1.0f;
#pragma unroll
    for (int i = 0; i < CHUNK; ++i) coefL *= COEF;

    // Phase B: wave-uniform reverse scan over later chunks' carries.
    float cin = 0.0f;
    for (int c = NCHUNK - 1; c > chunk; --c)
        cin = CARRY(c, lane) + coefL * cin;

    // Phase C: out[t] = local[t] + coef^(chunk_end - t) * cin ; single NT store.
    float* op = out + (size_t)t0 * B + col;
    float f = COEF;
#pragma unroll
    for (int i = CHUNK - 1; i >= 0; --i) {
        __builtin_nontemporal_store(local[i] + f * cin, op + (size_t)i * B);
        f *= COEF;
    }
#undef CARRY
}

// Generic fallback: one thread per column, serial over T. Correct for any shape.
__global__ void gae_generic(const float* __restrict__ rewards,
                            const float* __restrict__ values,
                            float* __restrict__ out, int T, int B)
{
    int col = blockIdx.x * blockDim.x + threadIdx.x;
    if (col >= B) return;
    float r = 0.0f;
    for (int t = T - 1; t >= 0; --t) {
        float d = rewards[(size_t)t * B + col] + DVAL * values[(size_t)(t + 1) * B + col];
        r = d + COEF * r;
        out[(size_t)t * B + col] = r;
    }
}

extern "C" void kernel_launch(void* const* d_in, const int* in_sizes, int n_in,
                              void* d_out, int out_size, void* d_ws, size_t ws_size,
                              hipStream_t stream)
{
    const float* rewards = (const float*)d_in[0];
    const float* values  = (const float*)d_in[1];
    float*       out     = (float*)d_out;

    // in_sizes[0] = (T+1)*B, out_size = T*B  =>  B = in_sizes[0] - out_size
    const int B = in_sizes[0] - out_size;
    const int T = (B > 0) ? (out_size / B) : 0;

    if (T == NCHUNK * CHUNK && B == 32768) {
        dim3 blk(COLS, NCHUNK, 1);      // 1024 threads = 32 wave32 waves
        dim3 grd(B / COLS, 1, 1);       // 1024 blocks -> 32768 waves total
        hipLaunchKernelGGL((gae_fast<32768>), grd, blk, 0, stream,
                           rewards, values, out);
    } else {
        const int th = 256;
        hipLaunchKernelGGL(gae_generic, dim3((B + th - 1) / th), dim3(th), 0, stream,
                           rewards, values, out, T, B);
    }
}